// Matrixfactorization_39350490366306
// MI455X (gfx1250) — compile-verified
//
#include <hip/hip_runtime.h>
#include <hip/hip_bf16.h>

// Matrix-factorization scoring: out[b] = dot(U[users[b]], F[feeds[b]]) + ub + fb + MU
// Mapped onto V_WMMA_F32_16X16X4_F32: per-wave 16-row tile, dot products appear
// on the diagonal of D = U_tile x F_tile^T accumulated over K=100 in steps of 4.

typedef __attribute__((ext_vector_type(2))) float v2f;
typedef __attribute__((ext_vector_type(8))) float v8f;

#define MF_DIM 100
#define MF_MU  0.5f

__global__ __launch_bounds__(256) void Matrixfactorization_39350490366306_kernel(
    const int*   __restrict__ users,
    const int*   __restrict__ feeds,
    const float* __restrict__ user_latent,
    const float* __restrict__ feed_latent,
    const float* __restrict__ user_bias,
    const float* __restrict__ feed_bias,
    float*       __restrict__ out,
    int batch)
{
    const int lane  = threadIdx.x & 31;
    const int wave  = threadIdx.x >> 5;
    const int tile  = blockIdx.x * 8 + wave;   // 16 batch rows per wave
    const int base  = tile * 16;
    if (base >= batch) return;                 // wave-uniform guard: EXEC stays all-1s

    // Lane L (L<16) serves row m=L with K-offsets {0,1}; lane L+16 serves the
    // same row with K-offsets {2,3}.  Identical layout for A (user) and B (feed).
    const int  m    = lane & 15;
    const int  b    = base + m;
    const int  koff = (lane >> 4) << 1;        // 0 or 2

    const long urow = (long)users[b] * MF_DIM;
    const long frow = (long)feeds[b] * MF_DIM;
    const float* __restrict__ up = user_latent + urow + koff;
    const float* __restrict__ fp = feed_latent + frow + koff;

    v8f acc = {};
#pragma unroll
    for (int k = 0; k < MF_DIM; k += 4) {
        v2f a = *(const v2f*)(up + k);         // A: 16x4 f32 slice (2 VGPRs)
        v2f c = *(const v2f*)(fp + k);         // B: 4x16 f32 slice (2 VGPRs)
        // v_wmma_f32_16x16x4_f32: D = A*B + C
        acc = __builtin_amdgcn_wmma_f32_16x16x4_f32(
            /*neg_a=*/false, a, /*neg_b=*/false, c,
            /*c_mod=*/(short)0, acc, /*reuse_a=*/false, /*reuse_b=*/false);
    }

    // Diagonal D[m][m] lives at: lanes 0..7  -> vgpr idx = lane   (m = lane)
    //                            lanes 24..31-> vgpr idx = lane-24 (m = lane-16)
    const int idx = (lane < 16) ? lane : (lane - 24);
    float d = acc[0];
    if (idx == 1) d = acc[1];
    if (idx == 2) d = acc[2];
    if (idx == 3) d = acc[3];
    if (idx == 4) d = acc[4];
    if (idx == 5) d = acc[5];
    if (idx == 6) d = acc[6];
    if (idx == 7) d = acc[7];

    const bool writer = (lane < 8) || (lane >= 24);
    if (writer) {
        out[b] = d + user_bias[b] + feed_bias[b] + MF_MU;
    }
}

extern "C" void kernel_launch(void* const* d_in, const int* in_sizes, int n_in,
                              void* d_out, int out_size, void* d_ws, size_t ws_size,
                              hipStream_t stream) {
    const int*   users       = (const int*)d_in[0];
    const int*   feeds       = (const int*)d_in[1];
    const float* user_latent = (const float*)d_in[2];
    const float* feed_latent = (const float*)d_in[3];
    const float* user_bias   = (const float*)d_in[4];
    const float* feed_bias   = (const float*)d_in[5];
    float*       out         = (float*)d_out;

    const int batch  = in_sizes[0];            // 16384
    const int waves  = (batch + 15) / 16;      // 16 rows per wave
    const int blocks = (waves + 7) / 8;        // 8 waves (256 threads) per block

    Matrixfactorization_39350490366306_kernel<<<blocks, 256, 0, stream>>>(
        users, feeds, user_latent, feed_latent, user_bias, feed_bias, out, batch);
}